// Net_63599875719659
// MI455X (gfx1250) — compile-verified
//
#include <hip/hip_runtime.h>

typedef __attribute__((ext_vector_type(2))) float v2f;
typedef __attribute__((ext_vector_type(8))) float v8f;

#define D_IN    64
#define D_HID   128
#define D_OUT   10
#define BATCH   512
#define N_TRAIN 10000
#define NCHUNKS 50
#define NCHUNK  200          // 50 * 200 == 10000
#define HSTRIDE 132          // 128 + 4 pad -> conflict-free LDS (132 mod 64 = 4)

// One wave per block. Computes 16 rows of  Z = relu(X @ W1^T) @ W2^T  -> [16, 10]
// using V_WMMA_F32_16X16X4_F32 for both GEMMs (fp32 inputs, full precision).
__global__ __launch_bounds__(32) void embed_wmma(
    const float* __restrict__ X, const float* __restrict__ W1,
    const float* __restrict__ W2, float* __restrict__ Zout)
{
    __shared__ float Hs[16 * HSTRIDE];
    const int lane = threadIdx.x & 31;
    const int m    = lane & 15;      // M (for A) / N (for B,C,D)
    const int hi   = lane >> 4;      // half-wave select
    const int kb   = hi * 2;         // f32 A/B operand K sub-offset (K=0,1 vs K=2,3)
    const int row0 = blockIdx.x * 16;

    // Preload lane's GEMM1 A operands: X[row0+m][4*ks + kb .. +1]  (shared by all 8 tiles)
    v2f a1[16];
    const float* xrow = X + (size_t)(row0 + m) * D_IN;
    #pragma unroll
    for (int ks = 0; ks < 16; ++ks)
        a1[ks] = *(const v2f*)(xrow + 4 * ks + kb);

    // GEMM1: H[16 x 128] = relu(X_tile @ W1^T), 8 column tiles of 16
    #pragma unroll
    for (int t = 0; t < 8; ++t) {
        v8f acc = {};
        // B[k][n] = W1^T[k][n] = W1[t*16+m][k] -> contiguous float2 along k
        const float* w1row = W1 + (size_t)(t * 16 + m) * D_IN;
        #pragma unroll
        for (int ks = 0; ks < 16; ++ks) {
            v2f b = *(const v2f*)(w1row + 4 * ks + kb);
            acc = __builtin_amdgcn_wmma_f32_16x16x4_f32(
                false, a1[ks], false, b, (short)0, acc, false, false);
        }
        #pragma unroll
        for (int v = 0; v < 8; ++v) {
            float r = acc[v] > 0.0f ? acc[v] : 0.0f;       // relu
            Hs[(v + hi * 8) * HSTRIDE + t * 16 + m] = r;   // D layout: row v+8*hi, col m
        }
    }
    __syncthreads();

    // GEMM2: Z[16 x 10] = H @ W2^T, N padded to 16 (cols 10..15 of B are zero)
    v8f acc2 = {};
    #pragma unroll
    for (int ks = 0; ks < 32; ++ks) {
        v2f a = *(const v2f*)(&Hs[m * HSTRIDE + 4 * ks + kb]);  // A: H[m][k..k+1]
        v2f b;
        if (m < D_OUT) {
            b = *(const v2f*)(W2 + (size_t)m * D_HID + 4 * ks + kb); // B[k][n]=W2[n][k]
        } else {
            b.x = 0.0f; b.y = 0.0f;
        }
        acc2 = __builtin_amdgcn_wmma_f32_16x16x4_f32(
            false, a, false, b, (short)0, acc2, false, false);
    }
    if (m < D_OUT) {
        #pragma unroll
        for (int v = 0; v < 8; ++v)
            Zout[(size_t)(row0 + v + hi * 8) * D_OUT + m] = acc2[v];
    }
}

// Grid: (NCHUNKS, BATCH/32). 320 threads = 10 waves; wave w owns output dim d=w,
// lane = batch index within 32-row b-tile. All lanes of a wave read the same LDS
// word -> broadcast. Deterministic partial sums (no atomics).
__global__ __launch_bounds__(320) void nw_partial(
    const float* __restrict__ Z, const float* __restrict__ Zw,
    const float* __restrict__ Y, const float* __restrict__ hptr,
    float* __restrict__ pnum, float* __restrict__ pden)
{
    __shared__ float Zs[NCHUNK * D_OUT];
    __shared__ float Ys[NCHUNK * D_OUT];
    const int tid = threadIdx.x;
    const int d   = tid >> 5;           // 0..9
    const int bl  = tid & 31;
    const int b   = blockIdx.y * 32 + bl;
    const int n0  = blockIdx.x * NCHUNK;

    for (int i = tid; i < NCHUNK * D_OUT; i += 320) {
        Zs[i] = Z[(size_t)n0 * D_OUT + i];
        Ys[i] = Y[(size_t)n0 * D_OUT + i];
    }
    __syncthreads();

    const float inv = 1.0f / hptr[0];
    const float zw  = Zw[(size_t)b * D_OUT + d];
    float num = 0.0f, den = 0.0f;
    #pragma unroll 4
    for (int i = 0; i < NCHUNK; ++i) {
        float u = (Zs[i * D_OUT + d] - zw) * inv;
        float e = __expf(-0.5f * u * u);        // v_exp_f32
        num = fmaf(e, Ys[i * D_OUT + d], num);
        den += e;
    }
    const size_t o = ((size_t)blockIdx.x * BATCH + b) * D_OUT + d;
    pnum[o] = num;
    pden[o] = den;
}

__global__ __launch_bounds__(256) void nw_reduce(
    const float* __restrict__ pnum, const float* __restrict__ pden,
    float* __restrict__ out)
{
    const int idx = blockIdx.x * 256 + threadIdx.x;  // 0 .. 5119
    if (idx >= BATCH * D_OUT) return;
    float num = 0.0f, den = 0.0f;
    #pragma unroll 5
    for (int c = 0; c < NCHUNKS; ++c) {
        num += pnum[(size_t)c * (BATCH * D_OUT) + idx];
        den += pden[(size_t)c * (BATCH * D_OUT) + idx];
    }
    out[idx] = num / den;
}

extern "C" void kernel_launch(void* const* d_in, const int* in_sizes, int n_in,
                              void* d_out, int out_size, void* d_ws, size_t ws_size,
                              hipStream_t stream) {
    const float* x  = (const float*)d_in[0];   // [512, 64]
    const float* tX = (const float*)d_in[1];   // [10000, 64]
    const float* Y  = (const float*)d_in[2];   // [10000, 10]
    const float* W1 = (const float*)d_in[3];   // [128, 64]
    const float* W2 = (const float*)d_in[4];   // [10, 128]
    const float* h  = (const float*)d_in[5];   // [1]
    float* out = (float*)d_out;                // [512, 10]

    float* ws = (float*)d_ws;
    float* Z   = ws;                                   // 100000 floats
    float* Zw  = Z  + (size_t)N_TRAIN * D_OUT;         //   5120 floats
    float* pn  = Zw + (size_t)BATCH * D_OUT;           // 256000 floats
    float* pd  = pn + (size_t)NCHUNKS * BATCH * D_OUT; // 256000 floats

    embed_wmma<<<N_TRAIN / 16, 32, 0, stream>>>(tX, W1, W2, Z);
    embed_wmma<<<BATCH / 16,  32, 0, stream>>>(x,  W1, W2, Zw);
    nw_partial<<<dim3(NCHUNKS, BATCH / 32), 320, 0, stream>>>(Z, Zw, Y, h, pn, pd);
    nw_reduce<<<(BATCH * D_OUT + 255) / 256, 256, 0, stream>>>(pn, pd, out);
}